// Kernel_8374174466554965253_77051713290583
// MI455X (gfx1250) — compile-verified
//
#include <hip/hip_runtime.h>
#include <hip/hip_bf16.h>
#include <math.h>

// MI455X / gfx1250, wave32. f32 WMMA 16x16x4 for all GEMM-like stages.
typedef float v2f __attribute__((ext_vector_type(2)));
typedef float v8f __attribute__((ext_vector_type(8)));

constexpr int NB = 8;          // batch
constexpr int CH = 128;        // channels
constexpr int HH = 112;
constexpr int WW = 112;
constexpr int HW = HH * WW;    // 12544 (sqrt = 112 exactly)

// fast sigmoid: v_exp + v_rcp (avoid IEEE divide expansion)
__device__ __forceinline__ float sigm(float v) {
    return __builtin_amdgcn_rcpf(1.0f + __expf(-v));
}

// ---------------- elementwise: t2 = silu(x) ----------------
__global__ void k_silu(const float* __restrict__ x, float* __restrict__ t2, int n) {
    int i = blockIdx.x * 256 + threadIdx.x;
    if (i < n) { float v = x[i]; t2[i] = v * sigm(v); }
}

// ---------------- wcomb[d, q=c'*7+k] = w15[d,q] * w10[32*(d>>5)+c', k] ----------------
__global__ void k_wcomb(const float* __restrict__ w15, const float* __restrict__ w10,
                        float* __restrict__ wc) {
    int i = blockIdx.x * 256 + threadIdx.x;
    if (i < CH * 224) {
        int d = i / 224, q = i - d * 224;
        int cp = q / 7, k = q - cp * 7;
        int g = d >> 5;
        wc[i] = w15[i] * w10[(32 * g + cp) * 7 + k];
    }
}

// ---------------- t5 split-K partials: part[wid][m*16+n] = sum_s sig(x[c,s]) * x[d,s] ----------------
// wid = ((nb*8 + ct)*8 + dt)*16 + ks ; K-chunk = 784 of 12544
__global__ void k_t5_partial(const float* __restrict__ x, float* __restrict__ part) {
    int wid = (blockIdx.x << 3) + (threadIdx.x >> 5);
    int lane = threadIdx.x & 31;
    int lm = lane & 15, lh = lane >> 4;
    int ks = wid & 15;
    int dt = (wid >> 4) & 7;
    int ct = (wid >> 7) & 7;
    int nb = wid >> 10;
    const float* xa = x + (size_t)(nb * CH + ct * 16 + lm) * HW + 2 * lh; // A rows: c
    const float* xb = x + (size_t)(nb * CH + dt * 16 + lm) * HW + 2 * lh; // B cols: d
    int s0 = ks * 784;
    v8f acc = {};
    for (int s = s0; s < s0 + 784; s += 4) {
        v2f av = *(const v2f*)(xa + s);   // K = {0,1} + 2*lh  (8B aligned)
        v2f b  = *(const v2f*)(xb + s);
        v2f a;
        a.x = sigm(av.x);
        a.y = sigm(av.y);
        acc = __builtin_amdgcn_wmma_f32_16x16x4_f32(false, a, false, b, (short)0, acc, false, false);
    }
    float* p = part + (size_t)wid * 256;
#pragma unroll
    for (int r = 0; r < 8; r++) p[(r + 8 * lh) * 16 + lm] = acc[r];
}

// ---------------- t9T[n,d,c] = (sum_ks part) * w9[c,d] / 112  (stored TRANSPOSED) ----------------
__global__ void k_t9(const float* __restrict__ part, const float* __restrict__ w9,
                     float* __restrict__ t9T) {
    int i = blockIdx.x * 256 + threadIdx.x;
    if (i >= NB * CH * CH) return;
    int d = i & 127;
    int c = (i >> 7) & 127;
    int nb = i >> 14;
    int ct = c >> 4, m = c & 15, dt = d >> 4, nn = d & 15;
    const float* p = part + (size_t)(((nb * 8 + ct) * 8 + dt) * 16) * 256 + m * 16 + nn;
    float s = 0.f;
#pragma unroll
    for (int ks = 0; ks < 16; ks++) s += p[ks * 256];
    t9T[((size_t)nb * CH + d) * CH + c] = s * w9[c * CH + d] * (1.0f / 112.0f);
}

// ---------------- t3: dilated 1x3 conv as 3 accumulated GEMMs (K=128 each) ----------------
// wid = (((nb*112 + h)*7 + wt)*8) + ot
__global__ void k_t3(const float* __restrict__ t2, const float* __restrict__ w3,
                     float* __restrict__ t3) {
    int wid = (blockIdx.x << 3) + (threadIdx.x >> 5);
    int lane = threadIdx.x & 31;
    int lm = lane & 15, lh = lane >> 4;
    int ot = wid & 7;
    int tmp = wid >> 3;
    int wt = tmp % 7; tmp /= 7;
    int h = tmp % HH;
    int nb = tmp / HH;
    int wcol = wt * 16 + lm;
    v8f acc = {};
    for (int j = 0; j < 3; j++) {
        int wsrc = wcol + 2 * (j - 1);
        bool ok = (wsrc >= 0) && (wsrc < WW);
        const float* bb = t2 + (size_t)nb * CH * HW + (size_t)h * WW + wsrc;
        const float* ab = w3 + (size_t)(ot * 16 + lm) * CH * 3 + j;
        for (int i0 = 0; i0 < CH; i0 += 4) {
            int k0 = i0 + 2 * lh;
            v2f a, b;
            a.x = ab[k0 * 3];
            a.y = ab[(k0 + 1) * 3];
            b.x = ok ? bb[(size_t)k0 * HW] : 0.f;
            b.y = ok ? bb[(size_t)(k0 + 1) * HW] : 0.f;
            acc = __builtin_amdgcn_wmma_f32_16x16x4_f32(false, a, false, b, (short)0, acc, false, false);
        }
    }
    float* o = t3 + (size_t)(nb * CH + ot * 16) * HW + (size_t)h * WW + wcol;
#pragma unroll
    for (int r = 0; r < 8; r++) o[(size_t)(r + 8 * lh) * HW] = acc[r];
}

// ---------------- t4: depthwise 7x1 dilated conv ----------------
__global__ void k_t4(const float* __restrict__ t3, const float* __restrict__ w4,
                     float* __restrict__ t4) {
    int i = blockIdx.x * 256 + threadIdx.x;
    if (i >= NB * CH * HW) return;
    int w = i % WW;
    int h = (i / WW) % HH;
    int c = (i / HW) % CH;
    size_t base = (size_t)(i / HW) * HW + w;
    float s = 0.f;
#pragma unroll
    for (int t = 0; t < 7; t++) {
        int hs = h - 9 + 3 * t;
        if (hs >= 0 && hs < HH) s += w4[c * 7 + t] * t3[base + (size_t)hs * WW];
    }
    t4[i] = s;
}

// ---------------- t14 GEMM: out[n,d,s] = sum_c t9T[n,d,c]*t2[n,c,s] / sqrt(128) ----------------
// wid = (nb*8 + dt)*784 + st
__global__ void k_t14(const float* __restrict__ t9T, const float* __restrict__ t2,
                      float* __restrict__ out) {
    int wid = (blockIdx.x << 3) + (threadIdx.x >> 5);
    int lane = threadIdx.x & 31;
    int lm = lane & 15, lh = lane >> 4;
    int st = wid % 784;
    int dt = (wid / 784) & 7;
    int nb = wid / (784 * 8);
    const float* a9 = t9T + ((size_t)nb * CH + dt * 16 + lm) * CH + 2 * lh; // contiguous in c
    const float* b2 = t2 + (size_t)nb * CH * HW + st * 16 + lm;             // + c*HW
    v8f acc = {};
    for (int c0 = 0; c0 < CH; c0 += 4) {
        int k0 = c0 + 2 * lh;
        v2f a = *(const v2f*)(a9 + c0);   // 8B-aligned pair
        v2f b;
        b.x = b2[(size_t)k0 * HW];
        b.y = b2[(size_t)(k0 + 1) * HW];
        acc = __builtin_amdgcn_wmma_f32_16x16x4_f32(false, a, false, b, (short)0, acc, false, false);
    }
    const float isc = 0.08838834764831845f; // 1/sqrt(128)
    float* o = out + (size_t)(nb * CH + dt * 16) * HW + st * 16 + lm;
#pragma unroll
    for (int r = 0; r < 8; r++) o[(size_t)(r + 8 * lh) * HW] = acc[r] * isc;
}

// ---------------- t15: fused unfold+rolls+grouped 1x1 conv as K=224 gather-GEMM on t4 ----------------
// out += sum_q wcomb[d,q] * t4[n,(32g+q/7-2)&127, ((h+2)%112)+2*(q%7)-6, (w-2)%112]
// wid = ((nb*8 + dt)*112 + h)*7 + wt
__global__ void k_t15(const float* __restrict__ t4, const float* __restrict__ wc,
                      float* __restrict__ out) {
    int wid = (blockIdx.x << 3) + (threadIdx.x >> 5);
    int lane = threadIdx.x & 31;
    int lm = lane & 15, lh = lane >> 4;
    int wt = wid % 7;
    int tmp = wid / 7;
    int h = tmp % HH; tmp /= HH;
    int dt = tmp & 7;
    int nb = tmp >> 3;
    int g = dt >> 1;
    int wcol = wt * 16 + lm;
    int wsrc = wcol - 2; if (wsrc < 0) wsrc += WW;
    int hh = h + 2; if (hh >= HH) hh -= HH;
    const float* ab = wc + (size_t)(dt * 16 + lm) * 224 + 2 * lh;
    const float* bb = t4 + (size_t)nb * CH * HW + wsrc;
    v8f acc = {};
    for (int q0 = 0; q0 < 224; q0 += 4) {
        int k0 = q0 + 2 * lh;
        v2f a = *(const v2f*)(ab + q0);   // contiguous 8B-aligned pair
        v2f b;
        {
            int q = k0, cp = q / 7, k = q - cp * 7;
            int cs = (g * 32 + cp - 2) & 127;
            int hs = hh + 2 * k - 6;
            b.x = (hs >= 0 && hs < HH) ? bb[(size_t)cs * HW + (size_t)hs * WW] : 0.f;
        }
        {
            int q = k0 + 1, cp = q / 7, k = q - cp * 7;
            int cs = (g * 32 + cp - 2) & 127;
            int hs = hh + 2 * k - 6;
            b.y = (hs >= 0 && hs < HH) ? bb[(size_t)cs * HW + (size_t)hs * WW] : 0.f;
        }
        acc = __builtin_amdgcn_wmma_f32_16x16x4_f32(false, a, false, b, (short)0, acc, false, false);
    }
    float* o = out + (size_t)(nb * CH + dt * 16) * HW + (size_t)h * WW + wcol;
#pragma unroll
    for (int r = 0; r < 8; r++) o[(size_t)(r + 8 * lh) * HW] += acc[r];
}

extern "C" void kernel_launch(void* const* d_in, const int* in_sizes, int n_in,
                              void* d_out, int out_size, void* d_ws, size_t ws_size,
                              hipStream_t stream) {
    const float* x   = (const float*)d_in[0];
    const float* w3  = (const float*)d_in[1];
    const float* w4  = (const float*)d_in[2];
    const float* w9  = (const float*)d_in[3];
    const float* w10 = (const float*)d_in[4];
    const float* w15 = (const float*)d_in[5];
    float* out = (float*)d_out;
    float* ws  = (float*)d_ws;

    const size_t NCHW = (size_t)NB * CH * HW;   // 12,845,056 floats
    float* t2   = ws;                           // 51.4 MB
    float* t3   = t2 + NCHW;                    // 51.4 MB
    float* t4   = t3 + NCHW;                    // 51.4 MB
    float* part = t4 + NCHW;                    // 8192 tiles * 256 = 8.4 MB
    float* t9T  = part + (size_t)8192 * 256;    // 0.5 MB (transposed [n,d,c])
    float* wcmb = t9T + (size_t)NB * CH * CH;   // 0.11 MB   (total ~164 MB)

    k_silu<<<(int)((NCHW + 255) / 256), 256, 0, stream>>>(x, t2, (int)NCHW);
    k_wcomb<<<112, 256, 0, stream>>>(w15, w10, wcmb);
    k_t5_partial<<<1024, 256, 0, stream>>>(x, part);          // 8192 waves
    k_t9<<<512, 256, 0, stream>>>(part, w9, t9T);
    k_t3<<<6272, 256, 0, stream>>>(t2, w3, t3);               // 50176 waves
    k_t4<<<50176, 256, 0, stream>>>(t3, w4, t4);
    k_t14<<<6272, 256, 0, stream>>>(t9T, t2, out);            // writes out
    k_t15<<<6272, 256, 0, stream>>>(t4, wcmb, out);           // accumulates into out
}